// LinearMultiShapeModel_2044404433113
// MI455X (gfx1250) — compile-verified
//
#include <hip/hip_runtime.h>

typedef __attribute__((ext_vector_type(2))) float v2f;
typedef __attribute__((ext_vector_type(8))) float v8f;

// One wave32 does everything:
//  Phase 0: X(6x4, padded 16x4) @ (m1+m2)(4x4, padded 4x16) via V_WMMA_F32_16X16X4_F32
//  Phase 1: faithful (x + x_) - x_ cancel pair (Linear w3/b3 + w4/b4)
//  Phase 2: per-row min (x2 branch) and mean (x3 branch)
//  Phase 3: the four output heads -> 23 floats
__global__ __launch_bounds__(32)
void lms_fused_kernel(const float* __restrict__ x,
                      const float* __restrict__ m1, const float* __restrict__ m2,
                      const float* __restrict__ m3, const float* __restrict__ m4,
                      const float* __restrict__ m5, const float* __restrict__ m6,
                      const float* __restrict__ m7, const float* __restrict__ m8,
                      const float* __restrict__ w1, const float* __restrict__ b1,
                      const float* __restrict__ w2, const float* __restrict__ b2,
                      const float* __restrict__ w3, const float* __restrict__ b3,
                      const float* __restrict__ w4, const float* __restrict__ b4,
                      float* __restrict__ out)
{
    __shared__ float xp[24];  // X' = X @ (m1+m2), 6 rows x 4 cols
    __shared__ float xf[24];  // x after the add/sub cancel pair
    __shared__ float v2s[6];  // min-reduced x2 branch
    __shared__ float v3s[6];  // mean-reduced x3 branch

    const int lane = threadIdx.x;     // 0..31, full wave
    const int m    = lane & 15;
    const int half = lane >> 4;

    // ---------------- Phase 0: WMMA f32 16x16x4 (EXEC all ones here) --------
    // A[m][k]: X rows 0..5 valid (x is (1,3,4,2) reshaped to (6,4): flat order)
    v2f a, b;
    a.x = (m < 6) ? x[m * 4 + (2 * half + 0)] : 0.0f;
    a.y = (m < 6) ? x[m * 4 + (2 * half + 1)] : 0.0f;
    // B[k][n]: (m1+m2), cols 0..3 valid, n = lane%16
    b.x = (m < 4) ? (m1[(2 * half + 0) * 4 + m] + m2[(2 * half + 0) * 4 + m]) : 0.0f;
    b.y = (m < 4) ? (m1[(2 * half + 1) * 4 + m] + m2[(2 * half + 1) * 4 + m]) : 0.0f;

    v8f c = {};
    v8f d = __builtin_amdgcn_wmma_f32_16x16x4_f32(
        /*neg_a=*/false, a, /*neg_b=*/false, b,
        /*c_mod=*/(short)0, c, /*reuse_a=*/false, /*reuse_b=*/false);

    // D[v + 8*half][lane%16]: rows 0..5 sit in VGPRs 0..5 of half==0 lanes.
    if (half == 0 && m < 4) {
#pragma unroll
        for (int r = 0; r < 6; ++r) xp[r * 4 + m] = d[r];
    }
    __syncthreads();

    // ---------------- Phase 1: x_ = Linear3(x) + Linear4(x); x = (x+x_)-x_ --
    if (lane < 24) {
        const int r = lane >> 2, cc = lane & 3;
        const float xs0 = xp[r * 4 + 0], xs1 = xp[r * 4 + 1];
        const float xs2 = xp[r * 4 + 2], xs3 = xp[r * 4 + 3];
        const float xv  = xp[r * 4 + cc];
        const float t3 = xs0 * w3[cc * 4 + 0] + xs1 * w3[cc * 4 + 1] +
                         xs2 * w3[cc * 4 + 2] + xs3 * w3[cc * 4 + 3] + b3[cc];
        const float t4 = xs0 * w4[cc * 4 + 0] + xs1 * w4[cc * 4 + 1] +
                         xs2 * w4[cc * 4 + 2] + xs3 * w4[cc * 4 + 3] + b4[cc];
        const float x_ = t3 + t4;
        xf[lane] = (xv + x_) - x_;
    }
    __syncthreads();

    // ---------------- Phase 2: per-row min (cols 0,1 / w1,b1) and mean ------
    if (lane < 6) {
        const int r = lane;
        const float a0 = xf[r * 4 + 0], a1 = xf[r * 4 + 1];   // x2 half
        const float c0 = xf[r * 4 + 2], c1 = xf[r * 4 + 3];   // x3 half
        float mn = INFINITY, sm = 0.0f;
#pragma unroll
        for (int j = 0; j < 8; ++j) {
            const float e2 = a0 * w1[j * 2 + 0] + a1 * w1[j * 2 + 1] + b1[j];
            mn = fminf(mn, e2);
            const float e3 = c0 * w2[j * 2 + 0] + c1 * w2[j * 2 + 1] + b2[j];
            sm += e3;
        }
        v2s[r] = mn;
        v3s[r] = sm * 0.125f;   // mean over 8
    }
    __syncthreads();

    // ---------------- Phase 3: output heads (23 floats total) ---------------
    if (lane < 6) {
        // x11: (1,3,8)@(8,2) -> flat (6,), scaled by m5[0]
        const int k = lane >> 1, cc = lane & 1;
        float acc = 0.0f;
#pragma unroll
        for (int j = 0; j < 8; ++j) acc += xf[k * 8 + j] * m3[j * 2 + cc];
        out[lane] = m5[0] * acc;
    } else if (lane == 6) {
        // x12: max over rows of (m4 @ x1), dotted with m6 -> scalar
        float acc = 0.0f;
#pragma unroll
        for (int j = 0; j < 8; ++j) {
            float cm = -INFINITY;
#pragma unroll
            for (int i = 0; i < 8; ++i) {
                const float s = m4[i * 3 + 0] * xf[0 * 8 + j] +
                                m4[i * 3 + 1] * xf[1 * 8 + j] +
                                m4[i * 3 + 2] * xf[2 * 8 + j];
                cm = fmaxf(cm, s);
            }
            acc += cm * m6[j];
        }
        out[6] = acc;
    } else if (lane < 13) {
        // x2 head: (6,) @ m7(6,6) -> (6,)
        const int cc = lane - 7;
        float acc = 0.0f;
#pragma unroll
        for (int r = 0; r < 6; ++r) acc += v2s[r] * m7[r * 6 + cc];
        out[7 + cc] = acc;
    } else if (lane < 23) {
        // x3 head: m8(10,6) @ (6,) -> (10,)
        const int i = lane - 13;
        float acc = 0.0f;
#pragma unroll
        for (int r = 0; r < 6; ++r) acc += m8[i * 6 + r] * v3s[r];
        out[13 + i] = acc;
    }
}

extern "C" void kernel_launch(void* const* d_in, const int* in_sizes, int n_in,
                              void* d_out, int out_size, void* d_ws, size_t ws_size,
                              hipStream_t stream) {
    (void)in_sizes; (void)n_in; (void)d_ws; (void)ws_size; (void)out_size;
    const float* x  = (const float*)d_in[0];
    const float* m1 = (const float*)d_in[1];
    const float* m2 = (const float*)d_in[2];
    const float* m3 = (const float*)d_in[3];
    const float* m4 = (const float*)d_in[4];
    const float* m5 = (const float*)d_in[5];
    const float* m6 = (const float*)d_in[6];
    const float* m7 = (const float*)d_in[7];
    const float* m8 = (const float*)d_in[8];
    const float* w1 = (const float*)d_in[9];
    const float* b1 = (const float*)d_in[10];
    const float* w2 = (const float*)d_in[11];
    const float* b2 = (const float*)d_in[12];
    const float* w3 = (const float*)d_in[13];
    const float* b3 = (const float*)d_in[14];
    const float* w4 = (const float*)d_in[15];
    const float* b4 = (const float*)d_in[16];
    float* out = (float*)d_out;

    lms_fused_kernel<<<1, 32, 0, stream>>>(x, m1, m2, m3, m4, m5, m6, m7, m8,
                                           w1, b1, w2, b2, w3, b3, w4, b4, out);
}